// TrussGNNEncoder_14379550507009
// MI455X (gfx1250) — compile-verified
//
#include <hip/hip_runtime.h>

typedef float v2f __attribute__((ext_vector_type(2)));
typedef float v8f __attribute__((ext_vector_type(8)));

__device__ __forceinline__ unsigned enc_ord(float f) {
  unsigned u = __float_as_uint(f);
  return (u & 0x80000000u) ? ~u : (u | 0x80000000u);
}
__device__ __forceinline__ float dec_ord(unsigned u) {
  return __uint_as_float((u & 0x80000000u) ? (u ^ 0x80000000u) : ~u);
}
__device__ __forceinline__ float leaky02(float x) { return x > 0.f ? x : 0.2f * x; }

// ---------------------------------------------------------------------------
// Fused GEMM: Hout[N,64] = X[N,64] @ W[64,64] (+bias), optional sout = Hout@avec
// One wave = 16 rows, 4 output tiles of 16x16, K accumulated with
// v_wmma_f32_16x16x4_f32 (16 k-steps).
// W staged in LDS pre-packed in B-fragment order:
//   Wl[p*64 + c] = { W[2p][c], W[2p+1][c] },  p = 2*kk + half
// so each B fragment is a single aligned ds_load_b64.
// ---------------------------------------------------------------------------
__global__ __launch_bounds__(256) void gemm64_wmma(
    const float* __restrict__ X, const float* __restrict__ W,
    const float* __restrict__ bias, float* __restrict__ Hout,
    const float* __restrict__ avec, float* __restrict__ sout, int N)
{
  __shared__ v2f Wl[32 * 64];
  __shared__ float al[64];
  __shared__ float bl[64];
  const int tid = threadIdx.x;
#pragma unroll
  for (int i = 0; i < 8; ++i) {
    int idx = tid + i * 256;            // 0..2047
    int p = idx >> 6, c = idx & 63;
    v2f w;
    w.x = W[(2 * p) * 64 + c];
    w.y = W[(2 * p + 1) * 64 + c];
    Wl[idx] = w;
  }
  if (tid < 64) {
    al[tid] = avec ? avec[tid] : 0.f;
    bl[tid] = bias ? bias[tid] : 0.f;
  }
  __syncthreads();

  const int wave = tid >> 5;
  const int lane = tid & 31;
  const int half = lane >> 4;   // lanes 0-15 vs 16-31
  const int lr   = lane & 15;
  const long long row0 = (long long)blockIdx.x * 128 + wave * 16;
  if (row0 >= N) return;        // wave-uniform (N % 16 == 0)

  // A fragment source: row = row0+lr, K = 4*kk + 2*half + {0,1}
  const float* xrow = X + (row0 + lr) * 64 + half * 2;

  const v8f vzero = {0.f, 0.f, 0.f, 0.f, 0.f, 0.f, 0.f, 0.f};
  v8f acc[4] = {vzero, vzero, vzero, vzero};

#pragma unroll
  for (int kk = 0; kk < 16; ++kk) {
    v2f a;
    a.x = xrow[kk * 4 + 0];
    a.y = xrow[kk * 4 + 1];
    const v2f* wp = &Wl[(kk * 2 + half) * 64 + lr];
#pragma unroll
    for (int n = 0; n < 4; ++n) {
      v2f b = wp[n * 16];
      acc[n] = __builtin_amdgcn_wmma_f32_16x16x4_f32(
          false, a, false, b, (short)0, acc[n], false, false);
    }
  }

  // bias (bl is zero when no bias)
#pragma unroll
  for (int n = 0; n < 4; ++n)
#pragma unroll
    for (int i = 0; i < 8; ++i)
      acc[n][i] += bl[n * 16 + lr];

  if (Hout) {
    // C layout: VGPR i -> M = i + 8*half, N = n*16 + lr
#pragma unroll
    for (int n = 0; n < 4; ++n)
#pragma unroll
      for (int i = 0; i < 8; ++i)
        Hout[(row0 + half * 8 + i) * 64 + n * 16 + lr] = acc[n][i];
  }

  if (sout) {
    float p[8];
#pragma unroll
    for (int i = 0; i < 8; ++i)
      p[i] = acc[0][i] * al[lr]       + acc[1][i] * al[16 + lr] +
             acc[2][i] * al[32 + lr]  + acc[3][i] * al[48 + lr];
#pragma unroll
    for (int m = 1; m <= 8; m <<= 1)
#pragma unroll
      for (int i = 0; i < 8; ++i)
        p[i] += __shfl_xor(p[i], m, 32);
    if (lr == 0) {
#pragma unroll
      for (int i = 0; i < 8; ++i)
        sout[row0 + half * 8 + i] = p[i];
    }
  }
}

// ---------------------------------------------------------------------------
// Encoders: out[N,64] = X[N,K] @ W[K,64] + b   (K = 6 or 4, too small for WMMA)
// ---------------------------------------------------------------------------
__global__ __launch_bounds__(256) void encode_kernel(
    const float* __restrict__ X, const float* __restrict__ W,
    const float* __restrict__ b, float* __restrict__ out, int N, int K)
{
  long long t = (long long)blockIdx.x * 256 + threadIdx.x;
  if (t >= (long long)N * 64) return;
  int n = (int)(t >> 6), j = (int)(t & 63);
  float s = b[j];
  for (int k = 0; k < K; ++k) s += X[(long long)n * K + k] * W[k * 64 + j];
  out[t] = s;
}

__global__ __launch_bounds__(256) void fill_f32(float* __restrict__ p, float v, long long n) {
  long long t = (long long)blockIdx.x * 256 + threadIdx.x;
  if (t < n) p[t] = v;
}
__global__ __launch_bounds__(256) void fill_u32(unsigned* __restrict__ p, unsigned v, long long n) {
  long long t = (long long)blockIdx.x * 256 + threadIdx.x;
  if (t < n) p[t] = v;
}

// ---------------------------------------------------------------------------
// Edge pass 1: segment-max of leaky_relu scores (both relations at once)
// ---------------------------------------------------------------------------
__global__ __launch_bounds__(256) void edge_pass1(
    const int* __restrict__ src_bar, const int* __restrict__ dst_jnt,
    const float* __restrict__ s_src_b, const float* __restrict__ s_dst_j,
    const float* __restrict__ s_src_j, const float* __restrict__ s_dst_b,
    unsigned* __restrict__ m_j, unsigned* __restrict__ m_b, int E)
{
  int e = blockIdx.x * 256 + threadIdx.x;
  if (e >= E) return;
  int sb = src_bar[e], dj = dst_jnt[e];
  float e1 = leaky02(s_src_b[sb] + s_dst_j[dj]);  // bar -> joint
  float e2 = leaky02(s_src_j[dj] + s_dst_b[sb]);  // joint -> bar
  atomicMax(&m_j[dj], enc_ord(e1));
  atomicMax(&m_b[sb], enc_ord(e2));
}

// ---------------------------------------------------------------------------
// Edge pass 2: ex = exp(e - max), segment-sum denominators
// ---------------------------------------------------------------------------
__global__ __launch_bounds__(256) void edge_pass2(
    const int* __restrict__ src_bar, const int* __restrict__ dst_jnt,
    const float* __restrict__ s_src_b, const float* __restrict__ s_dst_j,
    const float* __restrict__ s_src_j, const float* __restrict__ s_dst_b,
    const unsigned* __restrict__ m_j, const unsigned* __restrict__ m_b,
    float* __restrict__ den_j, float* __restrict__ den_b,
    float* __restrict__ ex_bj, float* __restrict__ ex_jb, int E)
{
  int e = blockIdx.x * 256 + threadIdx.x;
  if (e >= E) return;
  int sb = src_bar[e], dj = dst_jnt[e];
  float e1 = leaky02(s_src_b[sb] + s_dst_j[dj]);
  float e2 = leaky02(s_src_j[dj] + s_dst_b[sb]);
  float x1 = __expf(e1 - dec_ord(m_j[dj]));
  float x2 = __expf(e2 - dec_ord(m_b[sb]));
  ex_bj[e] = x1;
  ex_jb[e] = x2;
  atomicAdd(&den_j[dj], x1);
  atomicAdd(&den_b[sb], x2);
}

// ---------------------------------------------------------------------------
// Edge pass 3: out[dst] += alpha * h_src[src]; 64 threads per edge (coalesced)
// ---------------------------------------------------------------------------
__global__ __launch_bounds__(256) void edge_pass3(
    const int* __restrict__ src_bar, const int* __restrict__ dst_jnt,
    const float* __restrict__ ex_bj, const float* __restrict__ ex_jb,
    const float* __restrict__ den_j, const float* __restrict__ den_b,
    const float* __restrict__ hb_bj, const float* __restrict__ hj_jb,
    float* __restrict__ xj_new, float* __restrict__ xb_new, long long total)
{
  long long t = (long long)blockIdx.x * 256 + threadIdx.x;
  if (t >= total) return;
  long long e = t >> 6;
  int k = (int)(t & 63);
  int sb = src_bar[e], dj = dst_jnt[e];
  float a1 = ex_bj[e] / fmaxf(den_j[dj], 1e-16f);
  atomicAdd(&xj_new[(long long)dj * 64 + k], a1 * hb_bj[(long long)sb * 64 + k]);
  float a2 = ex_jb[e] / fmaxf(den_b[sb], 1e-16f);
  atomicAdd(&xb_new[(long long)sb * 64 + k], a2 * hj_jb[(long long)dj * 64 + k]);
}

__global__ __launch_bounds__(256) void bias_relu(
    float* __restrict__ x, const float* __restrict__ bias, long long total)
{
  long long t = (long long)blockIdx.x * 256 + threadIdx.x;
  if (t >= total) return;
  float v = x[t] + bias[t & 63];
  x[t] = v > 0.f ? v : 0.f;
}

// ---------------------------------------------------------------------------
// Critic pooling
// ---------------------------------------------------------------------------
__global__ __launch_bounds__(256) void pool_kernel(
    const float* __restrict__ xb, const int* __restrict__ batch,
    float* __restrict__ pooled, float* __restrict__ counts, long long total)
{
  long long t = (long long)blockIdx.x * 256 + threadIdx.x;
  if (t >= total) return;
  long long n = t >> 6;
  int k = (int)(t & 63);
  int g = batch[n];
  atomicAdd(&pooled[(long long)g * 64 + k], xb[t]);
  if (k == 0) atomicAdd(&counts[g], 1.0f);
}

__global__ __launch_bounds__(256) void pool_div(
    float* __restrict__ pooled, const float* __restrict__ counts, int total)
{
  int t = blockIdx.x * 256 + threadIdx.x;
  if (t >= total) return;
  pooled[t] /= fmaxf(counts[t >> 6], 1.0f);
}

// ---------------------------------------------------------------------------
// Head: act( relu(LN(row)*g+beta) @ w2 + b2 ), act 0=sigmoid 1=tanh
// ---------------------------------------------------------------------------
__global__ __launch_bounds__(256) void head_kernel(
    const float* __restrict__ T, const float* __restrict__ g,
    const float* __restrict__ beta, const float* __restrict__ w2,
    const float* __restrict__ b2, float* __restrict__ out, int N, int act)
{
  int n = blockIdx.x * 256 + threadIdx.x;
  if (n >= N) return;
  const float* r = T + (long long)n * 64;
  float mu = 0.f;
#pragma unroll
  for (int k = 0; k < 64; ++k) mu += r[k];
  mu *= (1.f / 64.f);
  float var = 0.f;
#pragma unroll
  for (int k = 0; k < 64; ++k) { float d = r[k] - mu; var += d * d; }
  var *= (1.f / 64.f);
  float inv = rsqrtf(var + 1e-5f);
  float dot = b2[0];
#pragma unroll
  for (int k = 0; k < 64; ++k) {
    float y = (r[k] - mu) * inv * g[k] + beta[k];
    y = y > 0.f ? y : 0.f;
    dot += y * w2[k];
  }
  out[n] = act ? tanhf(dot) : 1.f / (1.f + __expf(-dot));
}

// ---------------------------------------------------------------------------
extern "C" void kernel_launch(void* const* d_in, const int* in_sizes, int n_in,
                              void* d_out, int out_size, void* d_ws, size_t ws_size,
                              hipStream_t stream) {
  (void)n_in; (void)out_size; (void)ws_size;
  const float* joint_x  = (const float*)d_in[0];
  const float* bar_x    = (const float*)d_in[1];
  const int*   e_src_b  = (const int*)d_in[2];
  const int*   e_dst_j  = (const int*)d_in[3];
  const int*   barbatch = (const int*)d_in[4];
  const float* Wj = (const float*)d_in[5];   const float* bj = (const float*)d_in[6];
  const float* Wb = (const float*)d_in[7];   const float* bb = (const float*)d_in[8];
  const float* W_bj = (const float*)d_in[9]; const float* as_bj = (const float*)d_in[10];
  const float* ad_bj = (const float*)d_in[11]; const float* bias_bj = (const float*)d_in[12];
  const float* W_jb = (const float*)d_in[13]; const float* as_jb = (const float*)d_in[14];
  const float* ad_jb = (const float*)d_in[15]; const float* bias_jb = (const float*)d_in[16];
  const float* aW1 = (const float*)d_in[17]; const float* ab1 = (const float*)d_in[18];
  const float* ag  = (const float*)d_in[19]; const float* abeta = (const float*)d_in[20];
  const float* aW2 = (const float*)d_in[21]; const float* ab2 = (const float*)d_in[22];
  const float* cW1 = (const float*)d_in[23]; const float* cb1 = (const float*)d_in[24];
  const float* cg  = (const float*)d_in[25]; const float* cbeta = (const float*)d_in[26];
  const float* cW2 = (const float*)d_in[27]; const float* cb2 = (const float*)d_in[28];

  const int NJ = in_sizes[0] / 6;
  const int NB = in_sizes[1] / 4;
  const int E  = in_sizes[2];
  const int G  = 1024;

  char* wsp = (char*)d_ws;
  auto alloc = [&](size_t bytes) -> void* {
    void* p = (void*)wsp;
    wsp += (bytes + 255) & ~(size_t)255;
    return p;
  };
  float* xj[2] = { (float*)alloc((size_t)NJ * 64 * 4), (float*)alloc((size_t)NJ * 64 * 4) };
  float* xb[2] = { (float*)alloc((size_t)NB * 64 * 4), (float*)alloc((size_t)NB * 64 * 4) };
  float*    hb_bj  = (float*)alloc((size_t)NB * 64 * 4);
  float*    hj_jb  = (float*)alloc((size_t)NJ * 64 * 4);
  float*    s_src_b = (float*)alloc((size_t)NB * 4);
  float*    s_dst_j = (float*)alloc((size_t)NJ * 4);
  float*    s_src_j = (float*)alloc((size_t)NJ * 4);
  float*    s_dst_b = (float*)alloc((size_t)NB * 4);
  unsigned* m_j   = (unsigned*)alloc((size_t)NJ * 4);
  unsigned* m_b   = (unsigned*)alloc((size_t)NB * 4);
  float*    den_j = (float*)alloc((size_t)NJ * 4);
  float*    den_b = (float*)alloc((size_t)NB * 4);
  float*    ex_bj = (float*)alloc((size_t)E * 4);
  float*    ex_jb = (float*)alloc((size_t)E * 4);
  float*    pooled = (float*)alloc((size_t)G * 64 * 4);
  float*    counts = (float*)alloc((size_t)G * 4);
  float*    c1     = (float*)alloc((size_t)G * 64 * 4);

  auto blocks = [](long long n) { return (unsigned)((n + 255) / 256); };

  encode_kernel<<<blocks((long long)NJ * 64), 256, 0, stream>>>(joint_x, Wj, bj, xj[0], NJ, 6);
  encode_kernel<<<blocks((long long)NB * 64), 256, 0, stream>>>(bar_x, Wb, bb, xb[0], NB, 4);

  const unsigned ENC_NEG_INF = 0x007FFFFFu;  // enc_ord(-inf)
  int cur = 0;
  for (int l = 0; l < 3; ++l) {
    int nxt = cur ^ 1;
    const float* Wbj_l = W_bj + l * 4096;
    const float* Wjb_l = W_jb + l * 4096;
    unsigned gb = (unsigned)((NB + 127) / 128);
    unsigned gj = (unsigned)((NJ + 127) / 128);
    // src-side transforms need full H + score; dst-side only need the score.
    gemm64_wmma<<<gb, 256, 0, stream>>>(xb[cur], Wbj_l, nullptr, hb_bj, as_bj + l * 64, s_src_b, NB);
    gemm64_wmma<<<gj, 256, 0, stream>>>(xj[cur], Wbj_l, nullptr, nullptr, ad_bj + l * 64, s_dst_j, NJ);
    gemm64_wmma<<<gj, 256, 0, stream>>>(xj[cur], Wjb_l, nullptr, hj_jb, as_jb + l * 64, s_src_j, NJ);
    gemm64_wmma<<<gb, 256, 0, stream>>>(xb[cur], Wjb_l, nullptr, nullptr, ad_jb + l * 64, s_dst_b, NB);

    fill_u32<<<blocks(NJ), 256, 0, stream>>>(m_j, ENC_NEG_INF, NJ);
    fill_u32<<<blocks(NB), 256, 0, stream>>>(m_b, ENC_NEG_INF, NB);
    fill_f32<<<blocks(NJ), 256, 0, stream>>>(den_j, 0.f, NJ);
    fill_f32<<<blocks(NB), 256, 0, stream>>>(den_b, 0.f, NB);
    fill_f32<<<blocks((long long)NJ * 64), 256, 0, stream>>>(xj[nxt], 0.f, (long long)NJ * 64);
    fill_f32<<<blocks((long long)NB * 64), 256, 0, stream>>>(xb[nxt], 0.f, (long long)NB * 64);

    edge_pass1<<<blocks(E), 256, 0, stream>>>(e_src_b, e_dst_j, s_src_b, s_dst_j,
                                              s_src_j, s_dst_b, m_j, m_b, E);
    edge_pass2<<<blocks(E), 256, 0, stream>>>(e_src_b, e_dst_j, s_src_b, s_dst_j,
                                              s_src_j, s_dst_b, m_j, m_b,
                                              den_j, den_b, ex_bj, ex_jb, E);
    edge_pass3<<<blocks((long long)E * 64), 256, 0, stream>>>(
        e_src_b, e_dst_j, ex_bj, ex_jb, den_j, den_b,
        hb_bj, hj_jb, xj[nxt], xb[nxt], (long long)E * 64);

    bias_relu<<<blocks((long long)NJ * 64), 256, 0, stream>>>(xj[nxt], bias_bj + l * 64, (long long)NJ * 64);
    bias_relu<<<blocks((long long)NB * 64), 256, 0, stream>>>(xb[nxt], bias_jb + l * 64, (long long)NB * 64);
    cur = nxt;
  }

  float* probs = (float*)d_out;
  float* value = probs + NB;

  // Actor head
  float* t1 = hb_bj;  // layers done; reuse
  gemm64_wmma<<<(unsigned)((NB + 127) / 128), 256, 0, stream>>>(xb[cur], aW1, ab1, t1, nullptr, nullptr, NB);
  head_kernel<<<blocks(NB), 256, 0, stream>>>(t1, ag, abeta, aW2, ab2, probs, NB, 0);

  // Critic head
  fill_f32<<<blocks(G * 64), 256, 0, stream>>>(pooled, 0.f, (long long)G * 64);
  fill_f32<<<blocks(G), 256, 0, stream>>>(counts, 0.f, (long long)G);
  pool_kernel<<<blocks((long long)NB * 64), 256, 0, stream>>>(xb[cur], barbatch, pooled, counts, (long long)NB * 64);
  pool_div<<<blocks(G * 64), 256, 0, stream>>>(pooled, counts, G * 64);
  gemm64_wmma<<<(unsigned)((G + 127) / 128), 256, 0, stream>>>(pooled, cW1, cb1, c1, nullptr, nullptr, G);
  head_kernel<<<blocks(G), 256, 0, stream>>>(c1, cg, cbeta, cW2, cb2, value, G, 1);
}